// RTL_40492951667574
// MI455X (gfx1250) — compile-verified
//
#include <hip/hip_runtime.h>
#include <stdint.h>

// Problem constants (from the reference)
#define NUM_FEATURES 128
#define NUM_LATTICES 512
#define BATCH        16384
#define BT           32      // batch rows staged per workgroup (16 KB LDS tile)

typedef float f2v __attribute__((ext_vector_type(2)));
typedef float v8f __attribute__((ext_vector_type(8)));

// ---- CDNA5 async global->LDS DMA (ASYNCcnt path, CDNA5 ISA) ----
__device__ __forceinline__ void async_ld_b128(uint32_t lds_off, const void* gaddr) {
    asm volatile("global_load_async_to_lds_b128 %0, %1, off"
                 :: "v"(lds_off), "v"(gaddr)
                 : "memory");
}

__device__ __forceinline__ void wait_async0() {
#if __has_builtin(__builtin_amdgcn_s_wait_asynccnt)
    __builtin_amdgcn_s_wait_asynccnt(0);
#else
    asm volatile("s_wait_asynccnt 0" ::: "memory");
#endif
}

// ---------------------------------------------------------------------------
// Pre-pass: ws(16x512) = T(16x16) @ kern(16x512) on the matrix pipe.
//   T rows 0-7 : identity        -> ws[p,u]   = kern[p,u]          (klo)
//   T rows 8-15: K[p+8]-K[p]     -> ws[p+8,u] = kern[p+8,u]-kern[p,u] (kd)
// fp32 WMMA 16x16x4, 4 chained K-steps per 16-column tile; results are
// bit-exact (<=2 nonzero +/-1.0 products per output element).
// One wave per tile, 32 tiles. ISA 7.12.2 layouts:
//   A 16x4 : VGPR0={lanes0-15:K0, lanes16-31:K2}, VGPR1={K1, K3}, M=lane&15
//   B 4x16 : VGPR0={lanes0-15:row K0, lanes16-31:row K2}, VGPR1={K1, K3}, N=lane&15
//   C/D    : VGPR r = {lanes0-15: M=r, lanes16-31: M=r+8}, N=lane&15
// ---------------------------------------------------------------------------
__global__ __launch_bounds__(32)
void kprep_wmma(const float* __restrict__ kern, float* __restrict__ ws)
{
    const int lane = threadIdx.x;        // 0..31, EXEC all ones
    const int m    = lane & 15;          // M (for A) / N (for B, D)
    const int half = lane >> 4;          // K-pair selector
    const int NB   = blockIdx.x * 16;    // column tile base

    v8f acc = {};
#pragma unroll
    for (int j = 0; j < 4; ++j) {
        const int k0 = 4 * j + 2 * half;
        // A = T[m, k0], T[m, k0+1]
        f2v a;
        a.x = (float)(m == k0)     - (float)(m - 8 == k0);
        a.y = (float)(m == k0 + 1) - (float)(m - 8 == k0 + 1);
        // B = kern rows k0, k0+1 of this column tile
        f2v b;
        b.x = kern[k0 * NUM_LATTICES + NB + m];
        b.y = kern[(k0 + 1) * NUM_LATTICES + NB + m];
        acc = __builtin_amdgcn_wmma_f32_16x16x4_f32(
            /*neg_a=*/false, a, /*neg_b=*/false, b,
            /*c_mod=*/(short)0, acc, /*reuse_a=*/false, /*reuse_b=*/false);
    }
#pragma unroll
    for (int r = 0; r < 8; ++r)
        ws[(r + half * 8) * NUM_LATTICES + NB + m] = acc[r];
}

// VALU fallback pre-pass (same math, used only if ws path unavailable -> never
// expected; kept so kernel_launch is total).
__global__ __launch_bounds__(256)
void kprep_valu(const float* __restrict__ kern, float* __restrict__ ws)
{
    const int i = blockIdx.x * 256 + threadIdx.x;   // 0..8191
    if (i >= 16 * NUM_LATTICES) return;
    const int row = i / NUM_LATTICES, u = i % NUM_LATTICES;
    ws[i] = (row < 8) ? kern[i] : (kern[i] - kern[(row - 8) * NUM_LATTICES + u]);
}

// Multilinear interpolation of one lattice's 16 kernel values at (w0..w3).
// Identical algebra to the reference's product-expansion + dot product.
__device__ __forceinline__ float eval1(const float (&w)[4],
                                       const float (&klo)[8],
                                       const float (&kd)[8]) {
    float tv[8];
#pragma unroll
    for (int p = 0; p < 8; ++p)                 // lerp over bit3 (diffs precomputed)
        tv[p] = fmaf(w[3], kd[p], klo[p]);
    const float s0 = fmaf(w[2], tv[4] - tv[0], tv[0]);   // lerp over bit2
    const float s1 = fmaf(w[2], tv[5] - tv[1], tv[1]);
    const float s2 = fmaf(w[2], tv[6] - tv[2], tv[2]);
    const float s3 = fmaf(w[2], tv[7] - tv[3], tv[3]);
    const float r0 = fmaf(w[1], s2 - s0, s0);            // lerp over bit1
    const float r1 = fmaf(w[1], s3 - s1, s1);
    return fmaf(w[0], r1 - r0, r0);                      // lerp over bit0
}

// Main kernel: 256 threads, BT batch rows, all 512 lattices (2 consecutive
// lattices per thread -> b64 table loads and b64 output stores). The per-output
// contraction has no shared K dim (acc depends on b AND u via the random
// gather), so it stays on VALU; WMMA handles the kernel-table transform above.
__global__ __launch_bounds__(256)
void rtl_lerp_kernel(const float* __restrict__ x,
                     const int*   __restrict__ idx,
                     const float* __restrict__ ktab,   // transformed table (klo|kd)
                     float*       __restrict__ out)
{
    __shared__ float xs[BT * NUM_FEATURES];   // 16 KB x-tile

    const int t     = threadIdx.x;            // 0..255
    const int bBase = blockIdx.x * BT;

    // ---- kick off async DMA of the x tile: 4096 dwords = 4 x b128 per thread ----
    {
        const float* gsrc = x + (size_t)bBase * NUM_FEATURES + t * 16;
        const uint32_t loff = (uint32_t)(uintptr_t)(&xs[t * 16]);
#pragma unroll
        for (int i = 0; i < 4; ++i)
            async_ld_b128(loff + i * 16, gsrc + i * 4);
    }

    // ---- overlap with DMA: per-thread lattice state for u = 2t, 2t+1 ----
    const float* pw[2][4];                    // LDS base pointers xs + feature index
#pragma unroll
    for (int l = 0; l < 2; ++l) {
        const int4 iv = ((const int4*)idx)[2 * t + l];
        pw[l][0] = xs + iv.x;
        pw[l][1] = xs + iv.y;
        pw[l][2] = xs + iv.z;
        pw[l][3] = xs + iv.w;
    }

    float klo[2][8], kd[2][8];
#pragma unroll
    for (int p = 0; p < 8; ++p) {             // 16x b64, 256B/wave coalesced, no subs
        const f2v a = ((const f2v*)ktab)[p * (NUM_LATTICES / 2) + t];        // klo rows
        const f2v d = ((const f2v*)ktab)[(p + 8) * (NUM_LATTICES / 2) + t];  // kd  rows
        klo[0][p] = a.x; kd[0][p] = d.x;
        klo[1][p] = a.y; kd[1][p] = d.y;
    }

    wait_async0();       // own wave's async loads have written LDS
    __syncthreads();     // everyone's tile slice is visible

    // ---- software-pipelined main loop (prefetch distance 1) ----
    f2v* outp = (f2v*)(out + (size_t)bBase * NUM_LATTICES + 2 * t);

    float wc[2][4];
#pragma unroll
    for (int l = 0; l < 2; ++l)
#pragma unroll
        for (int d = 0; d < 4; ++d)
            wc[l][d] = __builtin_amdgcn_fmed3f(pw[l][d][0], 0.0f, 1.0f);

#pragma unroll 2
    for (int bl = 0; bl < BT - 1; ++bl) {
        float wn[2][4];
        const int ro = (bl + 1) * NUM_FEATURES;   // folds into ds_load imm offset
#pragma unroll
        for (int l = 0; l < 2; ++l)
#pragma unroll
            for (int d = 0; d < 4; ++d)
                wn[l][d] = __builtin_amdgcn_fmed3f(pw[l][d][ro], 0.0f, 1.0f);

        f2v o;
        o.x = eval1(wc[0], klo[0], kd[0]);
        o.y = eval1(wc[1], klo[1], kd[1]);
        __builtin_nontemporal_store(o, outp);     // streamed 32 MB output, th:NT
        outp += NUM_LATTICES / 2;

#pragma unroll
        for (int l = 0; l < 2; ++l)
#pragma unroll
            for (int d = 0; d < 4; ++d)
                wc[l][d] = wn[l][d];
    }

    f2v o;
    o.x = eval1(wc[0], klo[0], kd[0]);
    o.y = eval1(wc[1], klo[1], kd[1]);
    __builtin_nontemporal_store(o, outp);
}

extern "C" void kernel_launch(void* const* d_in, const int* in_sizes, int n_in,
                              void* d_out, int out_size, void* d_ws, size_t ws_size,
                              hipStream_t stream)
{
    (void)in_sizes; (void)n_in; (void)out_size;
    const float* x    = (const float*)d_in[0];   // (16384, 128) f32
    const int*   idx  = (const int*)  d_in[1];   // (512, 4) i32
    const float* kern = (const float*)d_in[2];   // (16, 512) f32
    float*       out  = (float*)d_out;           // (16384, 512) f32
    float*       ws   = (float*)d_ws;            // 16x512 transformed table (32 KB)

    if (ws_size >= (size_t)(16 * NUM_LATTICES * sizeof(float))) {
        kprep_wmma<<<NUM_LATTICES / 16, 32, 0, stream>>>(kern, ws);   // WMMA path
    } else {
        kprep_valu<<<(16 * NUM_LATTICES + 255) / 256, 256, 0, stream>>>(kern, ws);
    }
    rtl_lerp_kernel<<<BATCH / BT, 256, 0, stream>>>(x, idx, ws, out);
}